// GWNNLayer_67516885893333
// MI455X (gfx1250) — compile-verified
//
#include <hip/hip_runtime.h>

typedef float v2f __attribute__((ext_vector_type(2)));
typedef float v8f __attribute__((ext_vector_type(8)));
typedef unsigned int u32x4 __attribute__((ext_vector_type(4)));
typedef int i32x8 __attribute__((ext_vector_type(8)));
typedef int i32x4 __attribute__((ext_vector_type(4)));

#define NDIM 8192
#define KB 64                 // K-chunk per LDS tile (64 floats/row)
#define LDS_STRIDE (KB + 4)   // 68 dwords -> bank-conflict-free
#define WAVES 4               // waves per block
#define TILE_DW (16 * LDS_STRIDE)   // dwords per staged tile

#if defined(__has_builtin)
#if __has_builtin(__builtin_amdgcn_tensor_load_to_lds)
#define USE_TDM 1
#endif
#endif

// ---------------------------------------------------------------------------
// Kernel 1: T = features @ W   ([8192,16] @ [16,16])  -- trivial cost
// ---------------------------------------------------------------------------
__global__ __launch_bounds__(256) void feat_x_w(const float* __restrict__ F,
                                                const float* __restrict__ W,
                                                float* __restrict__ T) {
    int r = blockIdx.x * blockDim.x + threadIdx.x;
    if (r >= NDIM) return;
    float f[16];
#pragma unroll
    for (int k = 0; k < 16; ++k) f[k] = F[r * 16 + k];
#pragma unroll
    for (int c = 0; c < 16; ++c) {
        float acc = 0.f;
#pragma unroll
        for (int k = 0; k < 16; ++k) acc += f[k] * W[k * 16 + c];
        T[r * 16 + c] = acc;
    }
}

#ifdef USE_TDM
// Byte offset of a generic shared-memory pointer within the LDS segment.
__device__ __forceinline__ unsigned lds_byte_off(const void* p) {
    __attribute__((address_space(3))) const char* l =
        (__attribute__((address_space(3))) const char*)(const char*)p;
    return (unsigned)(unsigned long long)l;
}

// Issue TENSOR_LOAD_TO_LDS of a 16-row x KB-float 2D tile of an
// NDIM-stride fp32 tensor starting at gptr, into LDS byte offset ldsOff,
// with 4-dword padding after every 64-dword row (-> LDS_STRIDE layout).
__device__ __forceinline__ void tdm_load_tile(const float* gptr,
                                              unsigned ldsOff) {
    unsigned long long ga = (unsigned long long)(const void*)gptr;
    u32x4 g0;
    g0[0] = 1u;                                   // count=1, user descriptor
    g0[1] = ldsOff;                               // lds_addr [63:32]
    g0[2] = (unsigned)(ga & 0xFFFFFFFFu);         // global_addr lo
    g0[3] = (unsigned)((ga >> 32) & 0x01FFFFFFu)  // global_addr [56:32]
            | (2u << 30);                         // type=2 ("image")
    i32x8 g1;
    g1[0] = (2 << 16)      // data_size = 4 bytes
          | (1 << 20)      // pad_enable
          | (5 << 22)      // pad_interval: 64 dwords (one row)
          | (3 << 25);     // pad_amount: 4 dwords
    g1[1] = (int)((NDIM & 0xFFFF) << 16);               // tensor_dim0 lo16
    g1[2] = (int)((NDIM >> 16) | ((NDIM & 0xFFFF) << 16)); // dim0 hi / dim1 lo
    g1[3] = (int)((NDIM >> 16) | (KB << 16));           // dim1 hi / tile_dim0
    g1[4] = 16;                                         // tile_dim1=16 rows
    g1[5] = NDIM;                                       // tensor_dim0_stride lo
    g1[6] = 0;
    g1[7] = 0;
    i32x4 gz4 = {0, 0, 0, 0};           // groups 2/3 unused (2-D tensor)
    i32x8 gz8 = {0, 0, 0, 0, 0, 0, 0, 0};
    // amdgpu-toolchain (clang-23) 6-arg form.
    __builtin_amdgcn_tensor_load_to_lds(g0, g1, gz4, gz4, gz8, 0);
}
#endif

// ---------------------------------------------------------------------------
// C = A @ B, A:[8192,8192] fp32 row-major, B:[8192,16], C:[8192,16]
// Optional per-row output scale folds diag(filter) into the first GEMM.
// One wave per 16-row tile; V_WMMA_F32_16X16X4_F32 over K.
// TDM double-buffered LDS staging (fallback: coalesced b128 staging).
// ---------------------------------------------------------------------------
__global__ __launch_bounds__(WAVES * 32) void skinny_gemm_wmma(
    const float* __restrict__ A, const float* __restrict__ B,
    const float* __restrict__ scale, float* __restrict__ C) {
    __shared__ float lds[WAVES * 2 * TILE_DW];  // 2 tiles per wave

    const int lane = threadIdx.x & 31;
    const int wave = threadIdx.x >> 5;
    const int half = lane >> 4;   // 0: lanes 0-15, 1: lanes 16-31
    const int lid  = lane & 15;

    const int tile = blockIdx.x * WAVES + wave;
    const int row0 = tile * 16;

    float* buf0 = &lds[wave * 2 * TILE_DW];
    float* buf1 = buf0 + TILE_DW;

    v8f acc = {};  // 16x16 f32 accumulator tile (8 VGPRs)

    // 16 WMMA k-steps (K=4 each) against one staged tile.
    auto compute_chunk = [&](int chunk, const float* __restrict__ buf) {
        const float* bBase = B + (size_t)(chunk * KB + 2 * half) * 16 + lid;
#pragma unroll
        for (int kk = 0; kk < KB; kk += 4) {
            // A frag: lane holds row lid, K = {kk+2*half, kk+2*half+1}
            v2f a = *(const v2f*)(buf + lid * LDS_STRIDE + kk + 2 * half);
            // B frag: lane holds col lid, rows K and K+1
            v2f b;
            b.x = bBase[(size_t)kk * 16];
            b.y = bBase[(size_t)(kk + 1) * 16];
            acc = __builtin_amdgcn_wmma_f32_16x16x4_f32(
                false, a, false, b, (short)0, acc, false, false);
        }
    };

#ifdef USE_TDM
    const int nChunks = NDIM / KB;                     // 128
    const unsigned off0 = lds_byte_off(buf0);
    const unsigned off1 = lds_byte_off(buf1);
    const float* aRow = A + (size_t)row0 * NDIM;

    tdm_load_tile(aRow, off0);                         // prime chunk 0
    for (int i = 0; i < nChunks - 1; ++i) {
        // Previous reads of the target buffer must have drained (TDM is
        // unordered vs DS ops), then prefetch chunk i+1.
        asm volatile("s_wait_dscnt 0x0" ::: "memory");
        tdm_load_tile(aRow + (size_t)(i + 1) * KB, ((i + 1) & 1) ? off1 : off0);
        // TENSORcnt is in-order per wave: <=1 means chunk i has landed.
        asm volatile("s_wait_tensorcnt 0x1" ::: "memory");
        compute_chunk(i, (i & 1) ? buf1 : buf0);
    }
    asm volatile("s_wait_tensorcnt 0x0" ::: "memory");
    compute_chunk(nChunks - 1, ((nChunks - 1) & 1) ? buf1 : buf0);
#else
    // Fallback: synchronous coalesced b128 staging through registers.
    for (int k0 = 0; k0 < NDIM; k0 += KB) {
#pragma unroll
        for (int l = 0; l < (16 * KB) / 128; ++l) {    // 8 float4 per lane
            const int r = 2 * l + half;
            const int c = lid * 4;
            const float4 v =
                *(const float4*)(A + (size_t)(row0 + r) * NDIM + k0 + c);
            *(float4*)(buf0 + r * LDS_STRIDE + c) = v;
        }
        compute_chunk(k0 / KB, buf0);
    }
#endif

    // Epilogue: C/D layout -> acc[r] holds (row0 + r + 8*half, col lid)
    if (scale) {
#pragma unroll
        for (int r = 0; r < 8; ++r) acc[r] *= scale[row0 + r + 8 * half];
    }
#pragma unroll
    for (int r = 0; r < 8; ++r) {
        C[(size_t)(row0 + r + 8 * half) * 16 + lid] = acc[r];
    }
}

// ---------------------------------------------------------------------------
extern "C" void kernel_launch(void* const* d_in, const int* in_sizes, int n_in,
                              void* d_out, int out_size, void* d_ws,
                              size_t ws_size, hipStream_t stream) {
    (void)in_sizes; (void)n_in; (void)out_size; (void)ws_size;
    const float* features     = (const float*)d_in[0];
    const float* wavelets     = (const float*)d_in[1];
    const float* wavelets_inv = (const float*)d_in[2];
    const float* weight       = (const float*)d_in[3];
    const float* filt         = (const float*)d_in[4];
    float* out = (float*)d_out;

    float* T  = (float*)d_ws;        // [8192,16] transformed
    float* S2 = T + NDIM * 16;       // [8192,16] diag(filt) @ (Winv @ T)

    const int gemmBlocks = NDIM / (16 * WAVES);   // 128

    // 1) T = features @ W
    feat_x_w<<<NDIM / 256, 256, 0, stream>>>(features, weight, T);
    // 2) S2 = diag(filt) @ (wavelets_inv @ T)   (row-scale epilogue)
    skinny_gemm_wmma<<<gemmBlocks, WAVES * 32, 0, stream>>>(wavelets_inv, T,
                                                            filt, S2);
    // 3) out = wavelets @ S2
    skinny_gemm_wmma<<<gemmBlocks, WAVES * 32, 0, stream>>>(wavelets, S2,
                                                            nullptr, out);
}